// GCNConv_62182536511741
// MI455X (gfx1250) — compile-verified
//
#include <hip/hip_runtime.h>

typedef __attribute__((ext_vector_type(2)))  float v2f;
typedef __attribute__((ext_vector_type(8)))  float v8f;
typedef __attribute__((ext_vector_type(16))) int   i16v;
typedef __attribute__((ext_vector_type(16))) float f16v;

#define IN_DIM  256
#define OUT_DIM 128
#define EPW     256   // edges per wave in the SpMM kernel (multiple of 16)

// ---------------------------------------------------------------------------
// Kernel 0: zero the (poisoned) output buffer, float4 stores.
// ---------------------------------------------------------------------------
__global__ void zero_f4(float4* __restrict__ p, int n4) {
  int i = blockIdx.x * blockDim.x + threadIdx.x;
  if (i < n4) p[i] = make_float4(0.f, 0.f, 0.f, 0.f);
}

// ---------------------------------------------------------------------------
// Kernel 1: XW = X @ W via V_WMMA_F32_16X16X4_F32 (full f32 precision).
// grid.x = ceil(N/16); block = 256 threads = 8 waves.
// Wave w computes the 16x16 tile (mTile = blockIdx.x, nTile = w) of XW.
// Store epilogue uses one block-uniform bounds branch (no per-element EXEC
// masking); with N % 16 == 0 the unguarded path is always taken.
// ---------------------------------------------------------------------------
__global__ void __launch_bounds__(256)
gemm_wmma_f32(const float* __restrict__ X, const float* __restrict__ W,
              float* __restrict__ XW, int N) {
  const int lane   = threadIdx.x & 31;
  const int wave   = threadIdx.x >> 5;   // 0..7  -> N tile
  const int laneLo = lane & 15;
  const int laneHi = lane >> 4;          // 0 or 1
  const int mBase  = blockIdx.x * 16;

  int rowA = mBase + laneLo;
  if (rowA > N - 1) rowA = N - 1;        // clamp A loads for a ragged tail
  const int col = wave * 16 + laneLo;    // 0..127

  const float* __restrict__ Xr = X + (size_t)rowA * IN_DIM;

  v8f c = {};
#pragma unroll 8
  for (int k0 = 0; k0 < IN_DIM; k0 += 4) {
    const int k = k0 + 2 * laneHi;
    v2f a;
    a.x = Xr[k];
    a.y = Xr[k + 1];
    v2f b;
    b.x = W[(size_t)k       * OUT_DIM + col];
    b.y = W[(size_t)(k + 1) * OUT_DIM + col];
    // 8 args: (neg_a, A, neg_b, B, c_mod, C, reuse_a, reuse_b)
    c = __builtin_amdgcn_wmma_f32_16x16x4_f32(false, a, false, b,
                                              (short)0, c, false, false);
  }

  if (mBase + 16 <= N) {                 // block-uniform fast path
#pragma unroll
    for (int i = 0; i < 8; ++i)
      XW[(size_t)(mBase + i + 8 * laneHi) * OUT_DIM + col] = c[i];
  } else {
#pragma unroll
    for (int i = 0; i < 8; ++i) {
      const int r = mBase + i + 8 * laneHi;
      if (r < N) XW[(size_t)r * OUT_DIM + col] = c[i];
    }
  }
}

// ---------------------------------------------------------------------------
// Kernel 2: sorted-row segmented SpMM.
// One wave owns EPW consecutive edges; each lane owns 4 output columns
// (float4 -> global_load_b128 gather of the XW row, served from L2).
//
// Edge metadata is wave-uniform: anchoring the base index with
// readfirstlane lets the compiler issue scalar s_load_b512 of 16 edges'
// row/col/val into SGPRs (constant cache), so the inner loop has no
// shuffles and the row-change branch is a scalar branch.
//
// Flush policy (rows sorted): a segment that is neither the first nor the
// last of the wave's range is exclusively owned -> plain global_store_b128.
// Only the two boundary segments (which may be shared with neighbouring
// waves) use global_atomic_add_f32 into the zeroed output.
// ---------------------------------------------------------------------------
__global__ void __launch_bounds__(256)
spmm_seg(const float* __restrict__ XW, const int* __restrict__ erow,
         const int* __restrict__ ecol, const float* __restrict__ eval,
         float* __restrict__ out, int E) {
  const int lane = threadIdx.x & 31;
  const int wid  = blockIdx.x * (blockDim.x >> 5) + (threadIdx.x >> 5);

  const int e0 = __builtin_amdgcn_readfirstlane(wid * EPW);
  if (e0 >= E) return;
  int e1 = e0 + EPW;
  if (e1 > E) e1 = E;

  const int colBase = lane * 4;          // 32 lanes x 4 = 128 columns

  float4 acc    = make_float4(0.f, 0.f, 0.f, 0.f);
  int    curRow = -1;
  bool   firstSeg = true;                // first segment may leak into prev wave

  int eb = e0;
  // main path: 16-edge chunks via scalar loads
  for (; eb + 16 <= e1; eb += 16) {
    const i16v rv = *(const i16v*)(erow + eb);
    const i16v cv = *(const i16v*)(ecol + eb);
    const f16v vv = *(const f16v*)(eval + eb);
#pragma unroll
    for (int j = 0; j < 16; ++j) {
      const int   r = rv[j];
      const int   cI = cv[j];
      const float v = vv[j];
      if (r != curRow) {                 // scalar branch (uniform)
        if (curRow >= 0) {
          float* o = out + (size_t)curRow * OUT_DIM + colBase;
          if (firstSeg) {                // may be shared with previous wave
            atomicAdd(o + 0, acc.x); atomicAdd(o + 1, acc.y);
            atomicAdd(o + 2, acc.z); atomicAdd(o + 3, acc.w);
          } else {                       // exclusively owned -> b128 store
            *(float4*)o = acc;
          }
          firstSeg = false;
        }
        acc = make_float4(0.f, 0.f, 0.f, 0.f);
        curRow = r;
      }
      const float4 x = *(const float4*)(XW + (size_t)cI * OUT_DIM + colBase);
      acc.x = fmaf(v, x.x, acc.x);
      acc.y = fmaf(v, x.y, acc.y);
      acc.z = fmaf(v, x.z, acc.z);
      acc.w = fmaf(v, x.w, acc.w);
    }
  }
  // ragged tail (only the very last wave when E % 16 != 0)
  for (; eb < e1; ++eb) {
    const int   r = __builtin_amdgcn_readfirstlane(erow[eb]);
    const int   cI = __builtin_amdgcn_readfirstlane(ecol[eb]);
    const float v = __int_as_float(
        __builtin_amdgcn_readfirstlane(__float_as_int(eval[eb])));
    if (r != curRow) {
      if (curRow >= 0) {
        float* o = out + (size_t)curRow * OUT_DIM + colBase;
        if (firstSeg) {
          atomicAdd(o + 0, acc.x); atomicAdd(o + 1, acc.y);
          atomicAdd(o + 2, acc.z); atomicAdd(o + 3, acc.w);
        } else {
          *(float4*)o = acc;
        }
        firstSeg = false;
      }
      acc = make_float4(0.f, 0.f, 0.f, 0.f);
      curRow = r;
    }
    const float4 x = *(const float4*)(XW + (size_t)cI * OUT_DIM + colBase);
    acc.x = fmaf(v, x.x, acc.x);
    acc.y = fmaf(v, x.y, acc.y);
    acc.z = fmaf(v, x.z, acc.z);
    acc.w = fmaf(v, x.w, acc.w);
  }

  // final segment may extend into the next wave's range -> always atomic
  if (curRow >= 0) {
    float* o = out + (size_t)curRow * OUT_DIM + colBase;
    atomicAdd(o + 0, acc.x); atomicAdd(o + 1, acc.y);
    atomicAdd(o + 2, acc.z); atomicAdd(o + 3, acc.w);
  }
}

// ---------------------------------------------------------------------------
// Launcher. Inputs (flat): X [N*256] f32, W [256*128] f32,
// edge_row [E] i32, edge_col [E] i32, edge_val [E] f32.
// d_out: [N*128] f32. d_ws: XW scratch (N*128*4 = 51.2 MB).
// ---------------------------------------------------------------------------
extern "C" void kernel_launch(void* const* d_in, const int* in_sizes, int n_in,
                              void* d_out, int out_size, void* d_ws, size_t ws_size,
                              hipStream_t stream) {
  const float* X    = (const float*)d_in[0];
  const float* W    = (const float*)d_in[1];
  const int*   erow = (const int*)d_in[2];
  const int*   ecol = (const int*)d_in[3];
  const float* eval = (const float*)d_in[4];
  float*       out  = (float*)d_out;
  float*       XW   = (float*)d_ws;

  const int N = in_sizes[0] / IN_DIM;   // 100000
  const int E = in_sizes[2];            // 1600000

  // 1) zero output (it is poisoned before timing; rows without edges and
  //    atomic flushes both rely on a zeroed base)
  const int n4 = out_size / 4;
  zero_f4<<<(n4 + 255) / 256, 256, 0, stream>>>((float4*)out, n4);

  // 2) dense transform: XW = X @ W (WMMA f32)
  const int mTiles = (N + 15) / 16;
  gemm_wmma_f32<<<mTiles, 256, 0, stream>>>(X, W, XW, N);

  // 3) sorted-row segmented SpMM: scalar edge loads + boundary-only atomics
  const int wavesPerBlock = 256 / 32;                       // 8
  const int edgesPerBlock = wavesPerBlock * EPW;            // 2048
  const int blocks = (E + edgesPerBlock - 1) / edgesPerBlock;
  spmm_seg<<<blocks, 256, 0, stream>>>(XW, erow, ecol, eval, out, E);
}